// Transformerconv_55731495632941
// MI455X (gfx1250) — compile-verified
//
#include <hip/hip_runtime.h>
#include <hip/hip_bf16.h>
#include <math.h>
#include <stdint.h>

// CDNA5 / gfx1250: wave32, fp32 WMMA 16x16x4, TDM async tensor loads.
typedef __attribute__((ext_vector_type(2))) float        v2f;
typedef __attribute__((ext_vector_type(8))) float        v8f;
typedef __attribute__((ext_vector_type(4))) unsigned int v4u;
typedef __attribute__((ext_vector_type(4))) int          v4i;
typedef __attribute__((ext_vector_type(8))) int          v8i;

#define C_IN   128
#define C_OUT  64
#define EDGEF  64

// ---------------------------------------------------------------------------
// Issue a TDM 2-D tile load:  global (rows x rowlen f32, stride rowlen) -> LDS.
// D# packing per CDNA5 ISA (group0/group1; groups 2,3
// zero => 2-D tensor). Must be called by exactly one wave per tile.
// This toolchain exposes the 6-arg builtin:
//   (uint32x4 g0, int32x8 g1, int32x4 g2, int32x4 g3, int32x8, i32 cpol)
// ---------------------------------------------------------------------------
__device__ __forceinline__ void tdm_load_tile_f32(const float* gsrc, void* lds_dst,
                                                  unsigned rows, unsigned rowlen) {
    const uint64_t ga  = (uint64_t)(uintptr_t)gsrc;                 // byte address
    const uint32_t lds = (uint32_t)(uintptr_t)lds_dst;              // LDS byte offset (low 32b of aperture addr)

    v4u g0;
    g0[0] = 1u;                                                     // count=1, is_restore=0, no gather
    g0[1] = lds;                                                    // lds_addr
    g0[2] = (uint32_t)(ga & 0xFFFFFFFFu);                           // global_addr[31:0]
    g0[3] = (uint32_t)((ga >> 32) & 0x01FFFFFFu) | (2u << 30);      // global_addr[56:32] | type=2

    v8i g1;
    g1[0] = (int)(2u << 16);                                        // wg_mask=0, data_size=2 (4 bytes)
    g1[1] = (int)((rowlen & 0xFFFFu) << 16);                        // tensor_dim0[15:0] @ bits[63:48]
    g1[2] = (int)((rowlen >> 16) | ((rows & 0xFFFFu) << 16));       // tensor_dim0[31:16] | tensor_dim1[15:0]
    g1[3] = (int)((rows >> 16) | (rowlen << 16));                   // tensor_dim1[31:16] | tile_dim0
    g1[4] = (int)rows;                                              // tile_dim1 (tile_dim2=0)
    g1[5] = (int)rowlen;                                            // tensor_dim0_stride[31:0]
    g1[6] = 0;                                                      // stride hi | dim1_stride lo
    g1[7] = 0;

    const v4i z4 = {0, 0, 0, 0};
    const v8i z8 = {0, 0, 0, 0, 0, 0, 0, 0};
    __builtin_amdgcn_tensor_load_to_lds(g0, g1, z4, z4, z8, 0);
    __builtin_amdgcn_s_wait_tensorcnt((unsigned short)0);
}

// ---------------------------------------------------------------------------
// Float atomic max via signed/unsigned integer ordering trick (init = -inf).
// ---------------------------------------------------------------------------
__device__ __forceinline__ void atomicMaxF(float* addr, float val) {
    if (val >= 0.0f) {
        atomicMax((int*)addr, __float_as_int(val));
    } else {
        atomicMin((unsigned int*)addr, __float_as_uint(val));
    }
}

// ---------------------------------------------------------------------------
// Init kernel: amax = -inf, denom = 0 (per call; ws is not re-poisoned)
// ---------------------------------------------------------------------------
__global__ void init_segments(float* __restrict__ amax, float* __restrict__ denom, int n) {
    int i = blockIdx.x * blockDim.x + threadIdx.x;
    if (i < n) {
        amax[i]  = -INFINITY;
        denom[i] = 0.0f;
    }
}

// ---------------------------------------------------------------------------
// Kernel 1: fused node GEMM.  X[N x 128] @ {Wq,Wk,Wv,Wskip}[128 x 64].
// One block per 16-row tile; TDM stages the 16x128 A tile (8 KB) into LDS
// once; 8 waves each compute two 16x16 output tiles (16 col-tiles total).
// q,k,v -> ws; skip+bskip -> d_out (full deterministic overwrite).
// ---------------------------------------------------------------------------
__global__ void __launch_bounds__(256)
node_gemm_kernel(const float* __restrict__ X,
                 const float* __restrict__ Wq, const float* __restrict__ bq,
                 const float* __restrict__ Wk, const float* __restrict__ bk,
                 const float* __restrict__ Wv, const float* __restrict__ bv,
                 const float* __restrict__ Wskip, const float* __restrict__ bskip,
                 float* __restrict__ q, float* __restrict__ k, float* __restrict__ v,
                 float* __restrict__ outskip, int n_nodes) {
    __shared__ float sA[16 * C_IN];                   // 8 KB staged A tile
    const int row0 = blockIdx.x * 16;
    if (row0 + 16 > n_nodes) return;

    if (threadIdx.x < 32) {                           // wave 0 drives the TDM
        tdm_load_tile_f32(X + (size_t)row0 * C_IN, sA, 16u, (unsigned)C_IN);
    }
    __syncthreads();

    const int lane = threadIdx.x & 31;
    const int w    = threadIdx.x >> 5;
    // Fragment addressing per ISA VGPR layout tables (wave32):
    //  A 16x4: lanes 0-15 -> M=lane {K=ko,ko+1}; lanes 16-31 -> M=lane-16 {K=ko+2,ko+3}
    //  B 4x16: mirrored; C/D: VGPR r -> rows r / r+8, N = lane&15
    const int m  = lane & 15;
    const int ko = (lane >> 4) << 1;
    const float* Arow = &sA[m * C_IN];                // LDS reads -> ds_load

#pragma unroll
    for (int t = 0; t < 2; ++t) {
        const int ct = w * 2 + t;                     // 0..15
        const float* W; const float* bias; float* O;
        switch (ct >> 2) {
            case 0:  W = Wq;    bias = bq;    O = q;       break;
            case 1:  W = Wk;    bias = bk;    O = k;       break;
            case 2:  W = Wv;    bias = bv;    O = v;       break;
            default: W = Wskip; bias = bskip; O = outskip; break;
        }
        const int col0 = (ct & 3) * 16;

        v8f acc = {0.f, 0.f, 0.f, 0.f, 0.f, 0.f, 0.f, 0.f};
#pragma unroll
        for (int kk = 0; kk < C_IN; kk += 4) {
            v2f afrag, bfrag;
            afrag.x = Arow[kk + ko];
            afrag.y = Arow[kk + ko + 1];
            bfrag.x = W[(size_t)(kk + ko)     * C_OUT + col0 + m];
            bfrag.y = W[(size_t)(kk + ko + 1) * C_OUT + col0 + m];
            acc = __builtin_amdgcn_wmma_f32_16x16x4_f32(
                false, afrag, false, bfrag, (short)0, acc, false, false);
        }

        const float bb = bias[col0 + m];
        const int mbase = (lane >> 4) * 8;
#pragma unroll
        for (int r = 0; r < 8; ++r) {
            O[(size_t)(row0 + mbase + r) * C_OUT + col0 + m] = acc[r] + bb;
        }
    }
}

// ---------------------------------------------------------------------------
// Kernel 2: edge GEMM  e = edge_attr[E x 64] @ We[64 x 64] -> ws.
// One block per 32-row super-tile; TDM stages 32x64 f32 (8 KB) into LDS;
// 8 waves = 2 row-tiles x 4 col-tiles.
// ---------------------------------------------------------------------------
__global__ void __launch_bounds__(256)
edge_gemm_kernel(const float* __restrict__ EA, const float* __restrict__ We,
                 float* __restrict__ ebuf, int n_edges) {
    __shared__ float sA[32 * EDGEF];                  // 8 KB staged A super-tile
    const int row0 = blockIdx.x * 32;
    if (row0 + 32 > n_edges) return;

    if (threadIdx.x < 32) {
        tdm_load_tile_f32(EA + (size_t)row0 * EDGEF, sA, 32u, (unsigned)EDGEF);
    }
    __syncthreads();

    const int lane = threadIdx.x & 31;
    const int w    = threadIdx.x >> 5;
    const int rt   = w >> 2;                          // row tile within super-tile
    const int ct   = w & 3;                           // col tile
    const int col0 = ct * 16;

    const int m  = lane & 15;
    const int ko = (lane >> 4) << 1;
    const float* Arow = &sA[(rt * 16 + m) * EDGEF];

    v8f acc = {0.f, 0.f, 0.f, 0.f, 0.f, 0.f, 0.f, 0.f};
#pragma unroll
    for (int kk = 0; kk < EDGEF; kk += 4) {
        v2f afrag, bfrag;
        afrag.x = Arow[kk + ko];
        afrag.y = Arow[kk + ko + 1];
        bfrag.x = We[(size_t)(kk + ko)     * C_OUT + col0 + m];
        bfrag.y = We[(size_t)(kk + ko + 1) * C_OUT + col0 + m];
        acc = __builtin_amdgcn_wmma_f32_16x16x4_f32(
            false, afrag, false, bfrag, (short)0, acc, false, false);
    }

    const int rbase = row0 + rt * 16 + (lane >> 4) * 8;
#pragma unroll
    for (int r = 0; r < 8; ++r) {
        ebuf[(size_t)(rbase + r) * C_OUT + col0 + m] = acc[r];
    }
}

// ---------------------------------------------------------------------------
// Pass A: alpha[e] = dot(q[dst], k[src]+e) / sqrt(64); segment max -> amax.
// 16 lanes per edge, float4 loads, wave32 shfl_xor reduction.
// ---------------------------------------------------------------------------
__global__ void __launch_bounds__(256)
attn_alpha_kernel(const int* __restrict__ src, const int* __restrict__ dst,
                  const float* __restrict__ q, const float* __restrict__ k,
                  const float* __restrict__ ebuf,
                  float* __restrict__ alpha, float* __restrict__ amax, int n_edges) {
    const int t = blockIdx.x * blockDim.x + threadIdx.x;
    const int e = t >> 4;
    const int l = t & 15;
    if (e >= n_edges) return;
    const int s = src[e];
    const int d = dst[e];

    const float4 qv = ((const float4*)(q    + (size_t)d * C_OUT))[l];
    const float4 kv = ((const float4*)(k    + (size_t)s * C_OUT))[l];
    const float4 ev = ((const float4*)(ebuf + (size_t)e * C_OUT))[l];

    float dotv = qv.x * (kv.x + ev.x) + qv.y * (kv.y + ev.y)
               + qv.z * (kv.z + ev.z) + qv.w * (kv.w + ev.w);
#pragma unroll
    for (int msk = 8; msk >= 1; msk >>= 1) dotv += __shfl_xor(dotv, msk, 32);

    if (l == 0) {
        const float a = dotv * 0.125f;                // 1/sqrt(64)
        alpha[e] = a;
        atomicMaxF(amax + d, a);
    }
}

// ---------------------------------------------------------------------------
// Pass B: ex = exp(alpha - amax[dst]); alpha <- ex; denom[dst] += ex.
// (Reference's isfinite fixup only affects edge-less nodes, never indexed.)
// ---------------------------------------------------------------------------
__global__ void __launch_bounds__(256)
attn_softmax_kernel(const int* __restrict__ dst, float* __restrict__ alpha,
                    const float* __restrict__ amax, float* __restrict__ denom,
                    int n_edges) {
    const int e = blockIdx.x * blockDim.x + threadIdx.x;
    if (e >= n_edges) return;
    const int d = dst[e];
    const float ex = __expf(alpha[e] - amax[d]);
    alpha[e] = ex;
    atomicAdd(denom + d, ex);
}

// ---------------------------------------------------------------------------
// Pass C: out[dst] += (v[src] + e) * (ex / (denom[dst] + 1e-16)).
// 16 lanes per edge, 4 f32 global atomics per lane. d_out already holds skip.
// ---------------------------------------------------------------------------
__global__ void __launch_bounds__(256)
attn_scatter_kernel(const int* __restrict__ src, const int* __restrict__ dst,
                    const float* __restrict__ v, const float* __restrict__ ebuf,
                    const float* __restrict__ alpha, const float* __restrict__ denom,
                    float* __restrict__ out, int n_edges) {
    const int t = blockIdx.x * blockDim.x + threadIdx.x;
    const int e = t >> 4;
    const int l = t & 15;
    if (e >= n_edges) return;
    const int s = src[e];
    const int d = dst[e];

    const float w = alpha[e] / (denom[d] + 1e-16f);
    const float4 vv = ((const float4*)(v    + (size_t)s * C_OUT))[l];
    const float4 ev = ((const float4*)(ebuf + (size_t)e * C_OUT))[l];

    float* o = out + (size_t)d * C_OUT + l * 4;
    atomicAdd(o + 0, (vv.x + ev.x) * w);
    atomicAdd(o + 1, (vv.y + ev.y) * w);
    atomicAdd(o + 2, (vv.z + ev.z) * w);
    atomicAdd(o + 3, (vv.w + ev.w) * w);
}

// ---------------------------------------------------------------------------
// Launch
// ---------------------------------------------------------------------------
extern "C" void kernel_launch(void* const* d_in, const int* in_sizes, int n_in,
                              void* d_out, int out_size, void* d_ws, size_t ws_size,
                              hipStream_t stream) {
    const float* node_feats = (const float*)d_in[0];   // [N,128]
    const int*   edge_index = (const int*)d_in[1];     // [2,E]
    const float* edge_attr  = (const float*)d_in[2];   // [E,64]
    const float* Wq    = (const float*)d_in[3];
    const float* bq    = (const float*)d_in[4];
    const float* Wk    = (const float*)d_in[5];
    const float* bk    = (const float*)d_in[6];
    const float* Wv    = (const float*)d_in[7];
    const float* bv    = (const float*)d_in[8];
    const float* We    = (const float*)d_in[9];
    const float* Wskip = (const float*)d_in[10];
    const float* bskip = (const float*)d_in[11];
    float*       out   = (float*)d_out;

    const int N = in_sizes[0] / C_IN;      // 50000
    const int E = in_sizes[2] / EDGEF;     // 1200000
    const int* src = edge_index;
    const int* dst = edge_index + E;

    // ws layout (floats): q,k,v [N*64 each] | amax [N] | denom [N] | alpha [E] | ebuf [E*64]
    float* ws    = (float*)d_ws;
    float* q     = ws;
    float* k     = q + (size_t)N * C_OUT;
    float* v     = k + (size_t)N * C_OUT;
    float* amax  = v + (size_t)N * C_OUT;
    float* denom = amax + N;
    float* alpha = denom + N;
    float* ebuf  = alpha + E;

    // 0) per-call segment-state init
    init_segments<<<(N + 255) / 256, 256, 0, stream>>>(amax, denom, N);

    // 1) fused node GEMM: one block per 16-row tile
    node_gemm_kernel<<<(N + 15) / 16, 256, 0, stream>>>(
        node_feats, Wq, bq, Wk, bk, Wv, bv, Wskip, bskip, q, k, v, out, N);

    // 2) edge GEMM: one block per 32-row super-tile
    edge_gemm_kernel<<<(E + 31) / 32, 256, 0, stream>>>(edge_attr, We, ebuf, E);

    // 3) attention logits + segment max (16 lanes/edge)
    {
        const long long threads = (long long)E * 16;
        attn_alpha_kernel<<<(int)((threads + 255) / 256), 256, 0, stream>>>(
            src, dst, q, k, ebuf, alpha, amax, E);
    }

    // 4) exp + segment sum
    attn_softmax_kernel<<<(E + 255) / 256, 256, 0, stream>>>(dst, alpha, amax, denom, E);

    // 5) weighted scatter-add into out (already holds skip)
    {
        const long long threads = (long long)E * 16;
        attn_scatter_kernel<<<(int)((threads + 255) / 256), 256, 0, stream>>>(
            src, dst, v, ebuf, alpha, denom, out, E);
    }
}